// S3fdAssign_55697135894691
// MI455X (gfx1250) — compile-verified
//
#include <hip/hip_runtime.h>
#include <hip/hip_bf16.h>

#define POS_T 0.5f
#define NEG_T 0.3f
#define LOW_T 0.1f
#define MAX_GT 256
#define CHUNKS 8

typedef int v4i __attribute__((ext_vector_type(4)));
typedef __attribute__((address_space(1))) v4i* gv4i_p;   // global int4*
typedef __attribute__((address_space(3))) v4i* lv4i_p;   // LDS int4*

#if defined(__has_builtin)
#  if __has_builtin(__builtin_amdgcn_global_load_async_to_lds_b128)
#    define HAVE_ASYNC_LDS 1
#  endif
#endif

__device__ __forceinline__ void wait_async_zero() {
#if defined(__has_builtin)
#  if __has_builtin(__builtin_amdgcn_s_wait_asynccnt)
  __builtin_amdgcn_s_wait_asynccnt(0);
#  else
  asm volatile("s_wait_asynccnt 0" ::: "memory");
#  endif
#else
  asm volatile("s_wait_asynccnt 0" ::: "memory");
#endif
}

struct Top3 {
  float v0, v1, v2;
  int   i0, i1, i2;
};

__device__ __forceinline__ void top3_init(Top3& t) {
  t.v0 = t.v1 = t.v2 = -1.0f;
  t.i0 = t.i1 = t.i2 = -1;
}

__device__ __forceinline__ void top3_insert(Top3& t, float v, int i) {
  if (v > t.v2) {
    if (v > t.v0) {
      t.v2 = t.v1; t.i2 = t.i1;
      t.v1 = t.v0; t.i1 = t.i0;
      t.v0 = v;    t.i0 = i;
    } else if (v > t.v1) {
      t.v2 = t.v1; t.i2 = t.i1;
      t.v1 = v;    t.i1 = i;
    } else {
      t.v2 = v;    t.i2 = i;
    }
  }
}

__device__ __forceinline__ float iou_of(const float4 a, float areaA,
                                        const float4 g, float areaG) {
  float ltx = fmaxf(a.x, g.x);
  float lty = fmaxf(a.y, g.y);
  float rbx = fminf(a.z, g.z);
  float rby = fminf(a.w, g.w);
  float w = fmaxf(rbx - ltx, 0.0f);
  float h = fmaxf(rby - lty, 0.0f);
  float inter = w * h;
  // denom > 0 always (positive areas); fast v_rcp_f32 instead of a divide
  return inter * __builtin_amdgcn_rcpf(areaA + areaG - inter);
}

// ---------------------------------------------------------------------------
// Pass 1: per-anchor max/argmax over all GTs -> preliminary assignment.
// GT boxes staged into LDS with gfx1250 async global->LDS loads.
// Also initializes forced[] = -1 for the scatter-max pass.
// ---------------------------------------------------------------------------
__global__ __launch_bounds__(256) void s3fd_assign_pass(
    const float* __restrict__ anchor, const float* __restrict__ gt,
    float* __restrict__ out, int* __restrict__ forced, int n, int m) {
  __shared__ float4 sgt[MAX_GT];
  __shared__ float  sarea[MAX_GT];
  const int tid = threadIdx.x;

  if (tid < m) {
#ifdef HAVE_ASYNC_LDS
    __builtin_amdgcn_global_load_async_to_lds_b128(
        (gv4i_p)(gt + 4 * tid), (lv4i_p)(&sgt[tid]), 0, 0);
#else
    sgt[tid] = ((const float4*)gt)[tid];
#endif
  }
#ifdef HAVE_ASYNC_LDS
  wait_async_zero();
#endif
  __syncthreads();
  if (tid < m) {
    float4 g = sgt[tid];
    sarea[tid] = (g.z - g.x) * (g.w - g.y);
  }
  __syncthreads();

  const int i = blockIdx.x * 256 + tid;
  if (i >= n) return;

  // streamed once from HBM; give the prefetcher a hint (global_prefetch_b8)
  __builtin_prefetch(anchor + 4 * i + 4096, 0, 0);

  const float4 a = ((const float4*)anchor)[i];     // global_load_b128
  const float areaA = (a.z - a.x) * (a.w - a.y);

  float best = -1.0f;   // any IoU >= 0 beats it; '>' keeps first max (jax argmax)
  int bestIdx = 0;
#pragma unroll 4
  for (int j = 0; j < m; ++j) {
    float iou = iou_of(a, areaA, sgt[j], sarea[j]);
    if (iou > best) { best = iou; bestIdx = j; }
  }

  int assign = -2;
  if (best < NEG_T) assign = -1;
  if (best > POS_T) assign = bestIdx;
  out[i] = (float)assign;   // int32 values are exactly representable in f32
  forced[i] = -1;
}

// ---------------------------------------------------------------------------
// Pass 2a: per (gt, chunk) top-3 via per-thread top-3 + LDS tree merge.
// grid = m * CHUNKS blocks, 256 threads. Anchors are L2-resident (8 MB).
// ---------------------------------------------------------------------------
__global__ __launch_bounds__(256) void s3fd_topk_part(
    const float* __restrict__ anchor, const float* __restrict__ gt,
    float* __restrict__ cval, int* __restrict__ cidx, int n, int m) {
  const int tid   = threadIdx.x;
  const int gtIdx = blockIdx.x / CHUNKS;
  const int chunk = blockIdx.x % CHUNKS;
  const int len   = (n + CHUNKS - 1) / CHUNKS;
  const int lo    = chunk * len;
  const int hi    = min(n, lo + len);

  const float4 g = ((const float4*)gt)[gtIdx];
  const float areaG = (g.z - g.x) * (g.w - g.y);

  Top3 t;
  top3_init(t);
  for (int i = lo + tid; i < hi; i += 256) {
    const float4 a = ((const float4*)anchor)[i];
    const float areaA = (a.z - a.x) * (a.w - a.y);
    top3_insert(t, iou_of(a, areaA, g, areaG), i);
  }

  __shared__ float sv[256 * 3];
  __shared__ int   si[256 * 3];
  sv[tid * 3 + 0] = t.v0; sv[tid * 3 + 1] = t.v1; sv[tid * 3 + 2] = t.v2;
  si[tid * 3 + 0] = t.i0; si[tid * 3 + 1] = t.i1; si[tid * 3 + 2] = t.i2;
  __syncthreads();

  for (int s = 128; s > 0; s >>= 1) {
    if (tid < s) {
      Top3 u;
      u.v0 = sv[tid * 3 + 0]; u.v1 = sv[tid * 3 + 1]; u.v2 = sv[tid * 3 + 2];
      u.i0 = si[tid * 3 + 0]; u.i1 = si[tid * 3 + 1]; u.i2 = si[tid * 3 + 2];
      const int o = tid + s;
      top3_insert(u, sv[o * 3 + 0], si[o * 3 + 0]);
      top3_insert(u, sv[o * 3 + 1], si[o * 3 + 1]);
      top3_insert(u, sv[o * 3 + 2], si[o * 3 + 2]);
      sv[tid * 3 + 0] = u.v0; sv[tid * 3 + 1] = u.v1; sv[tid * 3 + 2] = u.v2;
      si[tid * 3 + 0] = u.i0; si[tid * 3 + 1] = u.i1; si[tid * 3 + 2] = u.i2;
    }
    __syncthreads();
  }

  if (tid == 0) {
    const int base = blockIdx.x * 3;
    cval[base + 0] = sv[0]; cval[base + 1] = sv[1]; cval[base + 2] = sv[2];
    cidx[base + 0] = si[0]; cidx[base + 1] = si[1]; cidx[base + 2] = si[2];
  }
}

// ---------------------------------------------------------------------------
// Pass 2b: one thread per GT merges its CHUNKS*3 candidates, then issues the
// claims. Reference resolves contested anchors by MAX claiming gt index ->
// atomicMax scatter reproduces that exactly.
// ---------------------------------------------------------------------------
__global__ __launch_bounds__(64) void s3fd_topk_final(
    const float* __restrict__ cval, const int* __restrict__ cidx,
    int* __restrict__ forced, int m) {
  const int mi = blockIdx.x * 64 + threadIdx.x;
  if (mi >= m) return;

  Top3 t;
  top3_init(t);
  for (int c = 0; c < CHUNKS; ++c) {
    const int base = (mi * CHUNKS + c) * 3;
    top3_insert(t, cval[base + 0], cidx[base + 0]);
    top3_insert(t, cval[base + 1], cidx[base + 1]);
    top3_insert(t, cval[base + 2], cidx[base + 2]);
  }

  const int npos = (t.v0 > POS_T) + (t.v1 > POS_T) + (t.v2 > POS_T);
  const bool low = npos < 3;

  if (t.i0 >= 0) atomicMax(&forced[t.i0], mi);   // k==0 always claims
  if (low) {
    if (t.i1 >= 0 && t.v1 > LOW_T) atomicMax(&forced[t.i1], mi);
    if (t.i2 >= 0 && t.v2 > LOW_T) atomicMax(&forced[t.i2], mi);
  }
}

// ---------------------------------------------------------------------------
// Pass 3: forced >= 0 overrides the preliminary assignment.
// ---------------------------------------------------------------------------
__global__ __launch_bounds__(256) void s3fd_finalize(
    const int* __restrict__ forced, float* __restrict__ out, int n) {
  const int i = blockIdx.x * 256 + threadIdx.x;
  if (i < n) {
    const int f = forced[i];
    if (f >= 0) out[i] = (float)f;
  }
}

extern "C" void kernel_launch(void* const* d_in, const int* in_sizes, int n_in,
                              void* d_out, int out_size, void* d_ws, size_t ws_size,
                              hipStream_t stream) {
  const float* anchor = (const float*)d_in[0];   // [N,4] f32
  const float* gt     = (const float*)d_in[1];   // [M,4] f32
  float* out = (float*)d_out;                    // [N] (int values stored as f32)
  const int n = in_sizes[0] / 4;
  const int m = in_sizes[1] / 4;

  // workspace layout: forced[n] int32 | cand_val[m*CHUNKS*3] f32 | cand_idx[...] i32
  int*   forced = (int*)d_ws;
  float* cval   = (float*)((char*)d_ws + (size_t)n * sizeof(int));
  int*   cidx   = (int*)((char*)cval + (size_t)m * CHUNKS * 3 * sizeof(float));

  const int blocks1 = (n + 255) / 256;
  s3fd_assign_pass<<<blocks1, 256, 0, stream>>>(anchor, gt, out, forced, n, m);
  s3fd_topk_part<<<m * CHUNKS, 256, 0, stream>>>(anchor, gt, cval, cidx, n, m);
  s3fd_topk_final<<<(m + 63) / 64, 64, 0, stream>>>(cval, cidx, forced, m);
  s3fd_finalize<<<blocks1, 256, 0, stream>>>(forced, out, n);
}